// InteractNet_85504208928872
// MI455X (gfx1250) — compile-verified
//
#include <hip/hip_runtime.h>
#include <math.h>

typedef __attribute__((ext_vector_type(2))) float v2f;
typedef __attribute__((ext_vector_type(8))) float v8f;

// ---------------------------------------------------------------- utilities
__global__ void k_zero(float* p, int n) {
  int i = blockIdx.x * 256 + threadIdx.x;
  int stride = gridDim.x * 256;
  for (; i < n; i += stride) p[i] = 0.f;
}

__global__ void k_deg(const int* __restrict__ dst, int E, float* __restrict__ deg) {
  int e = blockIdx.x * 256 + threadIdx.x;
  if (e < E) atomicAdd(&deg[dst[e]], 1.f);
}

__global__ void k_norm(float* deg, int N) {
  int i = blockIdx.x * 256 + threadIdx.x;
  if (i < N) {
    float d = deg[i];
    d = d < 1.f ? 1.f : d;
    deg[i] = rsqrtf(d);
  }
}

// copy a dense NxF block into a strided matrix at column offset
__global__ void k_copy_cols(const float* __restrict__ in, int ldin,
                            float* __restrict__ out, int ldout, int coff,
                            int N, int F) {
  int i = blockIdx.x * 256 + threadIdx.x;
  int tot = N * F, stride = gridDim.x * 256;
  for (; i < tot; i += stride) {
    int n = i / F, f = i - n * F;
    out[(size_t)n * ldout + coff + f] = in[(size_t)n * ldin + f];
  }
}

// zero-padded weight staging: Wp[c, k] = W[c, k] for c<Fout, k<K else 0
__global__ void k_padW(const float* __restrict__ W, int K, int Fout,
                       float* __restrict__ Wp, int Kpad, int cols) {
  int i = blockIdx.x * 256 + threadIdx.x;
  int tot = cols * Kpad, stride = gridDim.x * 256;
  for (; i < tot; i += stride) {
    int c = i / Kpad, k = i - c * Kpad;
    Wp[i] = (c < Fout && k < K) ? W[(size_t)c * K + k] : 0.f;
  }
}

// ------------------------------------------------- TAGConv hop: scatter-add
// agg[dst, f] += g[src, coff+f] * norm[src]   (one 32-lane row per edge)
__global__ void k_scatter(const int* __restrict__ src, const int* __restrict__ dst, int E,
                          const float* __restrict__ g, int ldg, int coff,
                          const float* __restrict__ norm, float* __restrict__ agg, int F) {
  int e = blockIdx.x * 8 + threadIdx.y;
  if (e >= E) return;
  int s = src[e], d = dst[e];
  float ns = norm[s];
  const float* gs = g + (size_t)s * ldg + coff;
  float* ad = agg + (size_t)d * F;
  for (int f = threadIdx.x; f < F; f += 32) atomicAdd(&ad[f], gs[f] * ns);
}

// fcat[n, coff+f] = agg[n,f] * norm[n]
__global__ void k_finish(float* __restrict__ fcat, int ld, int coff,
                         const float* __restrict__ agg, const float* __restrict__ norm,
                         int N, int F) {
  int i = blockIdx.x * 256 + threadIdx.x;
  int tot = N * F, stride = gridDim.x * 256;
  for (; i < tot; i += stride) {
    int n = i / F, f = i - n * F;
    fcat[(size_t)n * ld + coff + f] = agg[i] * norm[n];
  }
}

// -------------------------- TAGConv linear: Y = relu(X @ Wp^T + b) via WMMA
// X: Npad x Kpad (rows padded to mult of 64, Kpad mult of 4, pads zeroed).
// Wp: (NT*16) x Kpad zero-padded. Y: Npad x Fout.
// One wave per 16-row M tile, 4 waves/block, exact grid -> branch-free MMA
// loop with unconditional aligned b64 loads; EXEC all-ones at every WMMA.
template <int NT>
__global__ void k_linear_wmma(const float* __restrict__ X, int ldx,
                              const float* __restrict__ Wp,
                              const float* __restrict__ bias,
                              float* __restrict__ Y, int ldy,
                              int Kpad, int Fout) {
  int wave = threadIdx.x >> 5;
  int lane = threadIdx.x & 31;
  int mtile = blockIdx.x * 4 + wave;
  int mbase = mtile * 16;
  int half = lane >> 4;   // 0: holds k+0/k+1 ; 1: holds k+2/k+3 (ISA 16x4 f32 A layout)
  int l16 = lane & 15;
  int rowA = mbase + l16;

  const v2f* __restrict__ xp = (const v2f*)(X + (size_t)rowA * ldx);
  const v2f* __restrict__ wp[NT];
#pragma unroll
  for (int nt = 0; nt < NT; ++nt)
    wp[nt] = (const v2f*)(Wp + (size_t)(nt * 16 + l16) * Kpad);

  v8f acc[NT] = {};
  int kb2_end = Kpad >> 1;  // pair index; lane covers kb2 = half, half+2, ...
  for (int kb2 = half; kb2 < kb2_end; kb2 += 2) {
    v2f a = xp[kb2];
#pragma unroll
    for (int nt = 0; nt < NT; ++nt) {
      v2f bb = wp[nt][kb2];
      acc[nt] = __builtin_amdgcn_wmma_f32_16x16x4_f32(
          false, a, false, bb, (short)0, acc[nt], false, false);
    }
  }
#pragma unroll
  for (int nt = 0; nt < NT; ++nt) {
    int col = nt * 16 + l16;
    if (col >= Fout) continue;
    float bv = bias[col];
#pragma unroll
    for (int j = 0; j < 8; ++j) {
      int row = mbase + half * 8 + j;  // C/D layout: VGPR j -> M=j / M=j+8
      float v = acc[nt][j] + bv;
      Y[(size_t)row * ldy + col] = v > 0.f ? v : 0.f;
    }
  }
}

// ------------------------------------------ global attention pooling (F=31)
__global__ void k_gap(const float* __restrict__ feat, int ld, int N,
                      const float* __restrict__ w, const float* __restrict__ b,
                      float* __restrict__ out, float* __restrict__ sbuf) {
  __shared__ float red[256];
  __shared__ float Msh, Ssh;
  int tid = threadIdx.x;
  float lmax = -1e30f;
  for (int n = tid; n < N; n += 256) {
    float s = b[0];
    for (int f = 0; f < 31; ++f) s += feat[(size_t)n * ld + f] * w[f];
    sbuf[n] = s;
    lmax = fmaxf(lmax, s);
  }
  red[tid] = lmax; __syncthreads();
  for (int o = 128; o > 0; o >>= 1) { if (tid < o) red[tid] = fmaxf(red[tid], red[tid + o]); __syncthreads(); }
  if (tid == 0) Msh = red[0];
  __syncthreads();
  float M = Msh;
  float lsum = 0.f;
  for (int n = tid; n < N; n += 256) lsum += __expf(sbuf[n] - M);
  red[tid] = lsum; __syncthreads();
  for (int o = 128; o > 0; o >>= 1) { if (tid < o) red[tid] += red[tid + o]; __syncthreads(); }
  if (tid == 0) Ssh = red[0];
  __syncthreads();
  float Sinv = 1.f / Ssh;
  for (int f = 0; f < 31; ++f) {
    float lacc = 0.f;
    for (int n = tid; n < N; n += 256) lacc += __expf(sbuf[n] - M) * feat[(size_t)n * ld + f];
    red[tid] = lacc; __syncthreads();
    for (int o = 128; o > 0; o >>= 1) { if (tid < o) red[tid] += red[tid + o]; __syncthreads(); }
    if (tid == 0) out[f] = red[0] * Sinv;
    __syncthreads();
  }
}

// ------------------------------------------------------- bi-LSTM, H=31, T=140
__global__ void k_lstm(const float* __restrict__ seq,
                       const float* __restrict__ Wih_f, const float* __restrict__ Whh_f,
                       const float* __restrict__ bih_f, const float* __restrict__ bhh_f,
                       const float* __restrict__ Wih_b, const float* __restrict__ Whh_b,
                       const float* __restrict__ bih_b, const float* __restrict__ bhh_b,
                       float* __restrict__ xseq) {
  int dir = blockIdx.x;
  const float* Wih = dir ? Wih_b : Wih_f;
  const float* Whh = dir ? Whh_b : Whh_f;
  const float* bih = dir ? bih_b : bih_f;
  const float* bhh = dir ? bhh_b : bhh_f;
  __shared__ float h[32];
  __shared__ float xs[32];
  int j = threadIdx.x;
  h[j] = 0.f;
  float c = 0.f;
  __syncthreads();
  for (int t = 0; t < 140; ++t) {
    int tr = dir ? 139 - t : t;
    if (j < 31) xs[j] = seq[tr * 31 + j];
    __syncthreads();
    float hn = 0.f;
    if (j < 31) {
      float gi = bih[j] + bhh[j];
      float gf = bih[31 + j] + bhh[31 + j];
      float gg = bih[62 + j] + bhh[62 + j];
      float go = bih[93 + j] + bhh[93 + j];
      for (int k = 0; k < 31; ++k) {
        float xv = xs[k], hv = h[k];
        gi += Wih[j * 31 + k] * xv + Whh[j * 31 + k] * hv;
        gf += Wih[(31 + j) * 31 + k] * xv + Whh[(31 + j) * 31 + k] * hv;
        gg += Wih[(62 + j) * 31 + k] * xv + Whh[(62 + j) * 31 + k] * hv;
        go += Wih[(93 + j) * 31 + k] * xv + Whh[(93 + j) * 31 + k] * hv;
      }
      float iv = 1.f / (1.f + __expf(-gi));
      float fv = 1.f / (1.f + __expf(-gf));
      float gv = tanhf(gg);
      float ov = 1.f / (1.f + __expf(-go));
      c = fv * c + iv * gv;
      hn = ov * tanhf(c);
    }
    __syncthreads();
    if (j < 31) { h[j] = hn; xseq[tr * 62 + dir * 31 + j] = hn; }
    __syncthreads();
  }
}

// ------------------------------------------------------------------- MHA
__global__ void k_qkv(const float* __restrict__ x,
                      const float* __restrict__ Wq, const float* __restrict__ bq,
                      const float* __restrict__ Wk, const float* __restrict__ bk,
                      const float* __restrict__ Wv, const float* __restrict__ bv,
                      float* __restrict__ q, float* __restrict__ k, float* __restrict__ v) {
  int t = blockIdx.x, o = threadIdx.x;
  if (o >= 62) return;
  const float* xr = x + t * 62;
  float aq = bq[o], ak = bk[o], av = bv[o];
  for (int c = 0; c < 62; ++c) {
    float xv = xr[c];
    aq += Wq[o * 62 + c] * xv;
    ak += Wk[o * 62 + c] * xv;
    av += Wv[o * 62 + c] * xv;
  }
  q[t * 62 + o] = aq; k[t * 62 + o] = ak; v[t * 62 + o] = av;
}

__global__ void k_attn(const float* __restrict__ q, const float* __restrict__ k,
                       const float* __restrict__ v, float* __restrict__ o) {
  int t = blockIdx.x, h = blockIdx.y;
  __shared__ float sc[140];
  __shared__ float Msh, Ssh;
  int tid = threadIdx.x;  // 160 threads
  const float scale = 0.17960530202677491f;  // 1/sqrt(31)
  for (int kk = tid; kk < 140; kk += 160) {
    float s = 0.f;
    for (int d = 0; d < 31; ++d) s += q[t * 62 + h * 31 + d] * k[kk * 62 + h * 31 + d];
    s *= scale;
    // analytic MASK: visible iff (q==0&&k<=1) || (q==1&&k!=1) || (k==1&&q!=1)
    bool m = (t == 0 && kk <= 1) || (t == 1 && kk != 1) || (kk == 1 && t != 1);
    sc[kk] = m ? s : -1e9f;
  }
  __syncthreads();
  if (tid == 0) { float mx = -1e30f; for (int i = 0; i < 140; ++i) mx = fmaxf(mx, sc[i]); Msh = mx; }
  __syncthreads();
  for (int kk = tid; kk < 140; kk += 160) sc[kk] = __expf(sc[kk] - Msh);
  __syncthreads();
  if (tid == 0) { float su = 0.f; for (int i = 0; i < 140; ++i) su += sc[i]; Ssh = su; }
  __syncthreads();
  if (tid < 31) {
    float acc = 0.f;
    for (int kk = 0; kk < 140; ++kk) acc += sc[kk] * v[kk * 62 + h * 31 + tid];
    o[t * 62 + h * 31 + tid] = acc / Ssh;
  }
}

__global__ void k_proj(const float* __restrict__ x, const float* __restrict__ W,
                       const float* __restrict__ b, float* __restrict__ y) {
  int t = blockIdx.x, o = threadIdx.x;
  if (o >= 62) return;
  float a = b[o];
  for (int c = 0; c < 62; ++c) a += W[o * 62 + c] * x[t * 62 + c];
  y[t * 62 + o] = a;
}

// ---------------------------------------------------------------- FC layers
__global__ void k_fc1(const float* __restrict__ x, const float* __restrict__ W,
                      const float* __restrict__ b, float* __restrict__ y) {
  int o = blockIdx.x;
  const float* wr = W + (size_t)o * 8680;
  float acc = 0.f;
  for (int c = threadIdx.x; c < 8680; c += 256) {
    __builtin_prefetch(&wr[c + 2048], 0, 1);
    acc += x[c] * wr[c];
  }
  __shared__ float red[256];
  red[threadIdx.x] = acc; __syncthreads();
  for (int s = 128; s > 0; s >>= 1) { if (threadIdx.x < s) red[threadIdx.x] += red[threadIdx.x + s]; __syncthreads(); }
  if (threadIdx.x == 0) { float v = red[0] + b[o]; y[o] = v > 0.f ? v : 0.f; }
}

__global__ void k_fc2(const float* __restrict__ x, const float* __restrict__ W,
                      const float* __restrict__ b, float* __restrict__ out) {
  float acc = 0.f;
  for (int c = threadIdx.x; c < 4340; c += 256) acc += x[c] * W[c];
  __shared__ float red[256];
  red[threadIdx.x] = acc; __syncthreads();
  for (int s = 128; s > 0; s >>= 1) { if (threadIdx.x < s) red[threadIdx.x] += red[threadIdx.x + s]; __syncthreads(); }
  if (threadIdx.x == 0) out[0] = 1.f / (1.f + __expf(-(red[0] + b[0])));
}

static inline int hmin(int a, int b) { return a < b ? a : b; }
static inline int round_up(int v, int m) { return (v + m - 1) / m * m; }

extern "C" void kernel_launch(void* const* d_in, const int* in_sizes, int n_in,
                              void* d_out, int out_size, void* d_ws, size_t ws_size,
                              hipStream_t stream) {
  (void)n_in; (void)out_size; (void)ws_size;
  const float* feat_p = (const float*)d_in[0];
  const float* feat_l = (const float*)d_in[1];
  const int* edges_p = (const int*)d_in[2];
  const int* edges_l = (const int*)d_in[3];
  int Np = in_sizes[0] / 31;
  int Nl = in_sizes[1] / 74;
  int Ep = in_sizes[2] / 2;
  int El = in_sizes[3] / 2;
  const int* src_p = edges_p;  const int* dst_p = edges_p + Ep;
  const int* src_l = edges_l;  const int* dst_l = edges_l + El;

  int Np_pad = round_up(Np, 64);   // 64 rows = 4 waves x 16-row tiles (exact grids)
  int Nl_pad = round_up(Nl, 64);
  const int KP_P = 96;             // protein: K=93 -> pad 96, ld 96

  // workspace carve-up (f32)
  float* ws = (float*)d_ws;
  size_t off = 0;
  auto alloc = [&](size_t n) { float* p = ws + off; off += n; return p; };
  float* norm_p = alloc((size_t)Np);
  float* fcat_p = alloc((size_t)Np_pad * KP_P);  // [X|AX|A2X|pad], ld 96
  float* agg_p  = alloc((size_t)Np * 31);
  float* norm_l = alloc((size_t)Nl);
  float* fcat_l = alloc((size_t)Nl_pad * 224);   // ligand concat, ld up to 224
  float* featL  = alloc((size_t)Nl_pad * 74);    // ligand layer output
  float* agg_l  = alloc((size_t)Nl * 74);
  float* Wpad   = alloc((size_t)80 * 224);       // zero-padded weight staging
  float* sbuf   = alloc((size_t)Np);
  float* seq    = alloc(140 * 31);
  float* xseq   = alloc(140 * 62);
  float* qb     = alloc(140 * 62);
  float* kb     = alloc(140 * 62);
  float* vb     = alloc(140 * 62);
  float* ob     = alloc(140 * 62);
  float* aout   = alloc(140 * 62);
  float* hfc    = alloc(4340);

  auto zero = [&](float* p, size_t n) {
    int blocks = hmin((int)((n + 255) / 256), 4096);
    k_zero<<<blocks, 256, 0, stream>>>(p, (int)n);
  };

  // degree -> symmetric norm (in-degree per reference)
  zero(norm_p, (size_t)Np);
  k_deg<<<(Ep + 255) / 256, 256, 0, stream>>>(dst_p, Ep, norm_p);
  k_norm<<<(Np + 255) / 256, 256, 0, stream>>>(norm_p, Np);
  zero(norm_l, (size_t)Nl);
  k_deg<<<(El + 255) / 256, 256, 0, stream>>>(dst_l, El, norm_l);
  k_norm<<<(Nl + 255) / 256, 256, 0, stream>>>(norm_l, Nl);

  dim3 scat_block(32, 8);

  // ---------------- protein: 5 TAGConv layers, F=31 throughout (in-place)
  zero(fcat_p, (size_t)Np_pad * KP_P);   // pads (cols 93-95, rows >= Np) stay 0
  k_copy_cols<<<hmin((Np * 31 + 255) / 256, 4096), 256, 0, stream>>>(feat_p, 31, fcat_p, KP_P, 0, Np, 31);
  for (int layer = 0; layer < 5; ++layer) {
    const float* W = (const float*)d_in[4 + 2 * layer];
    const float* b = (const float*)d_in[5 + 2 * layer];
    for (int hop = 0; hop < 2; ++hop) {
      zero(agg_p, (size_t)Np * 31);
      k_scatter<<<(Ep + 7) / 8, scat_block, 0, stream>>>(src_p, dst_p, Ep, fcat_p, KP_P, hop * 31, norm_p, agg_p, 31);
      k_finish<<<hmin((Np * 31 + 255) / 256, 4096), 256, 0, stream>>>(fcat_p, KP_P, (hop + 1) * 31, agg_p, norm_p, Np, 31);
    }
    k_padW<<<hmin((32 * KP_P + 255) / 256, 4096), 256, 0, stream>>>(W, 93, 31, Wpad, KP_P, 32);
    // Y aliases fcat block0: safe (each wave reads only its own rows, stores last)
    k_linear_wmma<2><<<Np_pad / 64, 128, 0, stream>>>(fcat_p, KP_P, Wpad, b, fcat_p, KP_P, KP_P, 31);
  }

  // ---------------- ligand: 5 TAGConv layers, dims 74->70->65->60->55->31
  const int Fin[5] = {74, 70, 65, 60, 55};
  const int Fou[5] = {70, 65, 60, 55, 31};
  const float* lin = feat_l; int lin_ld = 74;
  for (int layer = 0; layer < 5; ++layer) {
    int F = Fin[layer], FO = Fou[layer];
    int K = 3 * F, Kpad = round_up(K, 4);
    int nt = (FO + 15) / 16, cols = nt * 16;
    zero(fcat_l, (size_t)Nl_pad * 224);  // ld changes per layer -> re-zero pads
    k_copy_cols<<<hmin((Nl * F + 255) / 256, 4096), 256, 0, stream>>>(lin, lin_ld, fcat_l, Kpad, 0, Nl, F);
    for (int hop = 0; hop < 2; ++hop) {
      zero(agg_l, (size_t)Nl * F);
      k_scatter<<<(El + 7) / 8, scat_block, 0, stream>>>(src_l, dst_l, El, fcat_l, Kpad, hop * F, norm_l, agg_l, F);
      k_finish<<<hmin((Nl * F + 255) / 256, 4096), 256, 0, stream>>>(fcat_l, Kpad, (hop + 1) * F, agg_l, norm_l, Nl, F);
    }
    const float* W = (const float*)d_in[14 + 2 * layer];
    const float* b = (const float*)d_in[15 + 2 * layer];
    k_padW<<<hmin((cols * Kpad + 255) / 256, 4096), 256, 0, stream>>>(W, K, FO, Wpad, Kpad, cols);
    if (nt <= 2)      k_linear_wmma<2><<<Nl_pad / 64, 128, 0, stream>>>(fcat_l, Kpad, Wpad, b, featL, FO, Kpad, FO);
    else if (nt == 4) k_linear_wmma<4><<<Nl_pad / 64, 128, 0, stream>>>(fcat_l, Kpad, Wpad, b, featL, FO, Kpad, FO);
    else              k_linear_wmma<5><<<Nl_pad / 64, 128, 0, stream>>>(fcat_l, Kpad, Wpad, b, featL, FO, Kpad, FO);
    lin = featL; lin_ld = FO;
  }

  // ---------------- pooling -> seq[140,31] (row0=ligand, row1=protein, rest 0)
  zero(seq, 140 * 31);
  k_gap<<<1, 256, 0, stream>>>(featL, 31, Nl, (const float*)d_in[26], (const float*)d_in[27], seq + 0, sbuf);
  k_gap<<<1, 256, 0, stream>>>(fcat_p, KP_P, Np, (const float*)d_in[24], (const float*)d_in[25], seq + 31, sbuf);

  // ---------------- bi-LSTM -> xseq[140,62]
  k_lstm<<<2, 32, 0, stream>>>(seq,
      (const float*)d_in[28], (const float*)d_in[29], (const float*)d_in[30], (const float*)d_in[31],
      (const float*)d_in[32], (const float*)d_in[33], (const float*)d_in[34], (const float*)d_in[35],
      xseq);

  // ---------------- MHA (2 heads, hd=31, analytic mask)
  k_qkv<<<140, 64, 0, stream>>>(xseq,
      (const float*)d_in[36], (const float*)d_in[37],
      (const float*)d_in[38], (const float*)d_in[39],
      (const float*)d_in[40], (const float*)d_in[41], qb, kb, vb);
  k_attn<<<dim3(140, 2), 160, 0, stream>>>(qb, kb, vb, ob);
  k_proj<<<140, 64, 0, stream>>>(ob, (const float*)d_in[42], (const float*)d_in[43], aout);

  // ---------------- FC head
  k_fc1<<<4340, 256, 0, stream>>>(aout, (const float*)d_in[44], (const float*)d_in[45], hfc);
  k_fc2<<<1, 256, 0, stream>>>(hfc, (const float*)d_in[46], (const float*)d_in[47], (float*)d_out);
}